// InvertedCognitionModel_86758339379779
// MI455X (gfx1250) — compile-verified
//
#include <hip/hip_runtime.h>
#include <math.h>

#define T_SEQ   2048
#define D_MODEL 768
#define D_FF    1536
#define BATCH   8
#define KQ_DIM  32
#define TOPK    4
#define ROWS    (BATCH * T_SEQ)   // 16384
#define S2_NBLK 24

typedef float v2f __attribute__((ext_vector_type(2)));
typedef float v8f __attribute__((ext_vector_type(8)));

__device__ __forceinline__ v8f wmma_f32(v2f a, v2f b, v8f c) {
  // V_WMMA_F32_16X16X4_F32: D = A(16x4) * B(4x16) + C, exact fp32
  return __builtin_amdgcn_wmma_f32_16x16x4_f32(false, a, false, b, (short)0, c,
                                               false, false);
}

__device__ __forceinline__ float gelu_exact(float v) {
  return 0.5f * v * (1.0f + erff(v * 0.70710678118654752f));
}

__device__ __forceinline__ v2f ld_b64(const float* p) {
  const float2 w = *(const float2*)p;   // p is 8B aligned (even k index)
  v2f r; r.x = w.x; r.y = w.y;
  return r;
}

// ---------------------------------------------------------------------------
// Tiled transpose: in (K x N) row-major -> out (N x K) row-major.
// Makes WMMA B-fragment loads contiguous along k (global_load_b64).
// ---------------------------------------------------------------------------
__global__ void __launch_bounds__(256)
transpose_kn(const float* __restrict__ in, float* __restrict__ out,
             int K, int N) {
  __shared__ float t[32][33];
  const int tx = threadIdx.x & 31;
  const int ty = threadIdx.x >> 5;            // 0..7
  const int n0 = blockIdx.x * 32;
  const int k0 = blockIdx.y * 32;
#pragma unroll
  for (int j = 0; j < 4; ++j)
    t[ty + j * 8][tx] = in[(size_t)(k0 + ty + j * 8) * N + n0 + tx];
  __syncthreads();
#pragma unroll
  for (int j = 0; j < 4; ++j)
    out[(size_t)(n0 + ty + j * 8) * K + k0 + tx] = t[tx][ty + j * 8];
}

// ---------------------------------------------------------------------------
// Router projections: q = x@qW + qb, k = x@kW + kb.  Block = 4 rows x 64 cols.
// ---------------------------------------------------------------------------
__global__ void router_proj(const float* __restrict__ x,
                            const float* __restrict__ qW, const float* __restrict__ qb,
                            const float* __restrict__ kW, const float* __restrict__ kb,
                            float* __restrict__ qo, float* __restrict__ ko) {
  const int tid  = threadIdx.x;
  const int rloc = tid >> 6;       // 0..3
  const int c    = tid & 63;
  const int j    = c & 31;
  const bool isK = (c >= 32);
  const int row  = blockIdx.x * 4 + rloc;
  const float* W = isK ? kW : qW;
  float s        = isK ? kb[j] : qb[j];
  const float* xr = x + (size_t)row * D_MODEL;
  for (int d = 0; d < D_MODEL; ++d) s += xr[d] * W[d * KQ_DIM + j];
  (isK ? ko : qo)[(size_t)row * KQ_DIM + j] = s;
}

// ---------------------------------------------------------------------------
// Router: per query row, 2048 scores (64/lane), 4x wave-argmax, gather-mean.
// ---------------------------------------------------------------------------
__global__ void __launch_bounds__(256)
router_topk_gather(const float* __restrict__ x, const float* __restrict__ qbuf,
                   const float* __restrict__ kbuf, float* __restrict__ x1) {
  const int lane = threadIdx.x & 31;
  const int wave = threadIdx.x >> 5;
  const int row  = blockIdx.x * 8 + wave;     // 0..16383
  const int b    = row >> 11;                 // / T_SEQ
  float qreg[KQ_DIM];
  const float* qr = qbuf + (size_t)row * KQ_DIM;
#pragma unroll
  for (int j = 0; j < KQ_DIM; ++j) qreg[j] = qr[j];

  float simv[64];
  const float* kb0 = kbuf + (size_t)b * T_SEQ * KQ_DIM;
#pragma unroll
  for (int i = 0; i < 64; ++i) {
    const float4* kr = (const float4*)(kb0 + (size_t)(i * 32 + lane) * KQ_DIM);
    float acc = 0.0f;
#pragma unroll
    for (int j4 = 0; j4 < KQ_DIM / 4; ++j4) {
      float4 kv = kr[j4];
      acc += qreg[4*j4+0]*kv.x + qreg[4*j4+1]*kv.y +
             qreg[4*j4+2]*kv.z + qreg[4*j4+3]*kv.w;
    }
    simv[i] = acc;   // 1/sqrt(KQ) scaling preserves ranking; skip
  }

  int chosen[TOPK];
#pragma unroll
  for (int r = 0; r < TOPK; ++r) chosen[r] = -1;
#pragma unroll
  for (int r = 0; r < TOPK; ++r) {
    float bv = -3.4e38f; int bi = 0x7fffffff;
#pragma unroll
    for (int i = 0; i < 64; ++i) {
      const int g = i * 32 + lane;
      bool skip = false;
#pragma unroll
      for (int rr = 0; rr < TOPK; ++rr) skip = skip || (g == chosen[rr]);
      const float v = skip ? -3.4e38f : simv[i];
      if (v > bv || (v == bv && g < bi)) { bv = v; bi = g; }
    }
#pragma unroll
    for (int off = 16; off > 0; off >>= 1) {
      const float ov = __shfl_xor(bv, off);
      const int   oi = __shfl_xor(bi, off);
      if (ov > bv || (ov == bv && oi < bi)) { bv = ov; bi = oi; }
    }
    chosen[r] = bi;   // identical in all lanes after butterfly
  }

  const float* xb = x + (size_t)b * T_SEQ * D_MODEL;
  const float* r0 = xb + (size_t)chosen[0] * D_MODEL;
  const float* r1 = xb + (size_t)chosen[1] * D_MODEL;
  const float* r2 = xb + (size_t)chosen[2] * D_MODEL;
  const float* r3 = xb + (size_t)chosen[3] * D_MODEL;
  float* o = x1 + (size_t)row * D_MODEL;
  for (int d = lane; d < D_MODEL; d += 32)
    o[d] = 0.25f * (r0[d] + r1[d] + r2[d] + r3[d]);
}

// ---------------------------------------------------------------------------
// System1 fused FFN, 16-row M-tile per block:
//   h = gelu(x1 @ f1W + f1b) in LDS (k-major), ypre = x1 + h @ f2W + f2b.
// Weights consumed from transposed copies (N x K): B frag = one b64 per wmma.
// Each wave runs 6 concurrent 16x16 accumulators to amortize A loads.
// ---------------------------------------------------------------------------
__global__ void __launch_bounds__(256)
system1_ffn(const float* __restrict__ x1, const float* __restrict__ f1WT,
            const float* __restrict__ f1b, const float* __restrict__ f2WT,
            const float* __restrict__ f2b, float* __restrict__ ypre) {
  extern __shared__ float lds[];
  float* hT  = lds;                 // [D_FF][16]    k-major
  float* x1T = lds + D_FF * 16;     // [D_MODEL][16] k-major
  const int tid  = threadIdx.x;
  const int lane = tid & 31;
  const int wave = tid >> 5;
  const int l15  = lane & 15;
  const int lh   = lane >> 4;
  const int row0 = blockIdx.x * 16;

  for (int idx = tid; idx < 16 * D_MODEL; idx += 256) {
    const int m = idx / D_MODEL, k = idx - m * D_MODEL;
    x1T[k * 16 + m] = x1[(size_t)(row0 + m) * D_MODEL + k];
  }
  __syncthreads();

  // GEMM1 + bias + gelu -> hT.  96 N-tiles: 2 groups x 6 tiles per wave.
  for (int g = 0; g < 2; ++g) {
    const float* pB[6];
    v8f acc[6];
#pragma unroll
    for (int c = 0; c < 6; ++c) {
      const int n0 = (wave * 12 + g * 6 + c) * 16;
      pB[c] = f1WT + (size_t)(n0 + l15) * D_MODEL;   // row of W^T, len 768
      __builtin_prefetch(pB[c], 0, 1);
      acc[c] = (v8f){0,0,0,0,0,0,0,0};
    }
    for (int k = 0; k < D_MODEL; k += 4) {
      const int kk = k + 2 * lh;
      v2f a;
      a.x = x1T[kk * 16 + l15];
      a.y = x1T[(kk + 1) * 16 + l15];
#pragma unroll
      for (int c = 0; c < 6; ++c)
        acc[c] = wmma_f32(a, ld_b64(pB[c] + kk), acc[c]);
    }
#pragma unroll
    for (int c = 0; c < 6; ++c) {
      const int n0 = (wave * 12 + g * 6 + c) * 16;
      const float bias = f1b[n0 + l15];
#pragma unroll
      for (int r = 0; r < 8; ++r) {
        const int mm = r + 8 * lh;
        hT[(n0 + l15) * 16 + mm] = gelu_exact(acc[c][r] + bias);
      }
    }
  }
  __syncthreads();

  // GEMM2 + bias + residual -> ypre. 48 N-tiles: 6 concurrent per wave, K=1536.
  {
    const float* pB[6];
    v8f acc[6];
#pragma unroll
    for (int c = 0; c < 6; ++c) {
      const int n0 = (wave * 6 + c) * 16;
      pB[c] = f2WT + (size_t)(n0 + l15) * D_FF;      // row of W^T, len 1536
      __builtin_prefetch(pB[c], 0, 1);
      acc[c] = (v8f){0,0,0,0,0,0,0,0};
    }
    for (int k = 0; k < D_FF; k += 4) {
      const int kk = k + 2 * lh;
      v2f a;
      a.x = hT[kk * 16 + l15];
      a.y = hT[(kk + 1) * 16 + l15];
#pragma unroll
      for (int c = 0; c < 6; ++c)
        acc[c] = wmma_f32(a, ld_b64(pB[c] + kk), acc[c]);
    }
#pragma unroll
    for (int c = 0; c < 6; ++c) {
      const int n0 = (wave * 6 + c) * 16;
      const float bias = f2b[n0 + l15];
#pragma unroll
      for (int r = 0; r < 8; ++r) {
        const int mm = r + 8 * lh;
        const float v = acc[c][r] + bias + x1T[(n0 + l15) * 16 + mm];
        ypre[(size_t)(row0 + mm) * D_MODEL + n0 + l15] = v;
      }
    }
  }
}

// ---------------------------------------------------------------------------
// LayerNorm in place: one wave per 768-wide row.
// ---------------------------------------------------------------------------
__global__ void __launch_bounds__(256)
layernorm_rows(float* __restrict__ y, const float* __restrict__ g,
               const float* __restrict__ bta) {
  const int lane = threadIdx.x & 31;
  const int wave = threadIdx.x >> 5;
  const int row  = blockIdx.x * 8 + wave;
  float* yr = y + (size_t)row * D_MODEL;
  float v[24];
  float s = 0.0f;
#pragma unroll
  for (int i = 0; i < 24; ++i) { v[i] = yr[i * 32 + lane]; s += v[i]; }
#pragma unroll
  for (int off = 16; off > 0; off >>= 1) s += __shfl_xor(s, off);
  const float mu = s * (1.0f / (float)D_MODEL);
  float q = 0.0f;
#pragma unroll
  for (int i = 0; i < 24; ++i) { const float d = v[i] - mu; q += d * d; }
#pragma unroll
  for (int off = 16; off > 0; off >>= 1) q += __shfl_xor(q, off);
  const float inv = rsqrtf(q * (1.0f / (float)D_MODEL) + 1e-5f);
#pragma unroll
  for (int i = 0; i < 24; ++i) {
    const int c = i * 32 + lane;
    yr[c] = (v[i] - mu) * inv * g[c] + bta[c];
  }
}

// ---------------------------------------------------------------------------
// Grid-wide barrier: monotone-epoch atomic counter (zeroed each launch).
// ---------------------------------------------------------------------------
__device__ __forceinline__ void grid_barrier(unsigned* cnt, unsigned& epoch,
                                             unsigned nblk) {
  __threadfence();
  __syncthreads();
  if (threadIdx.x == 0) {
    atomicAdd(cnt, 1u);
    ++epoch;
    const unsigned target = epoch * nblk;
    while (atomicAdd(cnt, 0u) < target) { __builtin_amdgcn_s_sleep(8); }
  }
  __syncthreads();
  __threadfence();
}

// ---------------------------------------------------------------------------
// System2 persistent scan: 2048 sequential steps, M=8 padded to 16.
// Phase1: a = gelu(ei @ t1W + t1b)   96 tiles, 4 concurrent per block.
// Phase2: prop = a @ t2W + t2b; mem = mem*(1-sig)+prop*sig  48 tiles, 2/blk.
// 16x16 tile K=1536 split across the block's 8 waves; LDS tree reduce.
// ---------------------------------------------------------------------------
__global__ void __launch_bounds__(256)
system2_scan(const float* __restrict__ y,
             const float* __restrict__ t1WT, const float* __restrict__ t1b,
             const float* __restrict__ t2WT, const float* __restrict__ t2b,
             float* __restrict__ a2, float* __restrict__ mem,
             unsigned* __restrict__ cnt) {
  __shared__ float eiT[D_FF * 8];    // 48KB: A staged k-major (8 live rows)
  __shared__ float red[8 * 256];     // 8KB : cross-wave partial reduce
  const int tid  = threadIdx.x;
  const int lane = tid & 31;
  const int wave = tid >> 5;
  const int l15  = lane & 15;
  const int lh   = lane >> 4;
  const int blk  = blockIdx.x;
  unsigned epoch = 0;

  for (int t = 0; t < T_SEQ; ++t) {
    // stage ei^T = concat(y[:,t,:], mem)^T
    for (int idx = tid; idx < 8 * D_FF; idx += 256) {
      const int mrow = idx / D_FF;
      const int kk   = idx - mrow * D_FF;
      const float v  = (kk < D_MODEL)
                         ? y[((size_t)mrow * T_SEQ + t) * D_MODEL + kk]
                         : mem[mrow * D_MODEL + (kk - D_MODEL)];
      eiT[kk * 8 + mrow] = v;
    }
    __syncthreads();

    // ---- phase 1: 4 concurrent tiles ----
    {
      const int kbase = wave * 192;
      const float* pB[4];
      v8f acc[4];
#pragma unroll
      for (int c = 0; c < 4; ++c) {
        const int n0 = (blk * 4 + c) * 16;
        pB[c] = t1WT + (size_t)(n0 + l15) * D_FF;
        acc[c] = (v8f){0,0,0,0,0,0,0,0};
      }
      for (int k = 0; k < 192; k += 4) {
        const int kk = kbase + k + 2 * lh;
        v2f a;
        a.x = (l15 < 8) ? eiT[kk * 8 + l15] : 0.0f;
        a.y = (l15 < 8) ? eiT[(kk + 1) * 8 + l15] : 0.0f;
#pragma unroll
        for (int c = 0; c < 4; ++c)
          acc[c] = wmma_f32(a, ld_b64(pB[c] + kk), acc[c]);
      }
#pragma unroll
      for (int c = 0; c < 4; ++c) {
        const int n0 = (blk * 4 + c) * 16;
#pragma unroll
        for (int r = 0; r < 8; ++r) red[wave * 256 + r * 32 + lane] = acc[c][r];
        __syncthreads();
        float s = 0.0f;
#pragma unroll
        for (int w = 0; w < 8; ++w) s += red[w * 256 + tid];
        const int rr = tid >> 5, ll = tid & 31;
        const int mm = rr + 8 * (ll >> 4), nn = ll & 15;
        if (mm < 8) a2[mm * D_FF + n0 + nn] = gelu_exact(s + t1b[n0 + nn]);
        __syncthreads();
      }
    }
    grid_barrier(cnt, epoch, S2_NBLK);

    // stage a^T
    for (int idx = tid; idx < 8 * D_FF; idx += 256) {
      const int mrow = idx / D_FF;
      const int kk   = idx - mrow * D_FF;
      eiT[kk * 8 + mrow] = a2[mrow * D_FF + kk];
    }
    __syncthreads();

    // ---- phase 2: 2 concurrent tiles ----
    {
      const int kbase = wave * 192;
      const float* pB[2];
      v8f acc[2];
#pragma unroll
      for (int c = 0; c < 2; ++c) {
        const int n0 = (blk * 2 + c) * 16;
        pB[c] = t2WT + (size_t)(n0 + l15) * D_FF;
        acc[c] = (v8f){0,0,0,0,0,0,0,0};
      }
      for (int k = 0; k < 192; k += 4) {
        const int kk = kbase + k + 2 * lh;
        v2f a;
        a.x = (l15 < 8) ? eiT[kk * 8 + l15] : 0.0f;
        a.y = (l15 < 8) ? eiT[(kk + 1) * 8 + l15] : 0.0f;
#pragma unroll
        for (int c = 0; c < 2; ++c)
          acc[c] = wmma_f32(a, ld_b64(pB[c] + kk), acc[c]);
      }
#pragma unroll
      for (int c = 0; c < 2; ++c) {
        const int n0 = (blk * 2 + c) * 16;
#pragma unroll
        for (int r = 0; r < 8; ++r) red[wave * 256 + r * 32 + lane] = acc[c][r];
        __syncthreads();
        float s = 0.0f;
#pragma unroll
        for (int w = 0; w < 8; ++w) s += red[w * 256 + tid];
        const int rr = tid >> 5, ll = tid & 31;
        const int mm = rr + 8 * (ll >> 4), nn = ll & 15;
        if (mm < 8) {
          const float prop = s + t2b[n0 + nn];
          const float g    = 1.0f / (1.0f + expf(-prop));
          const float mo   = mem[mm * D_MODEL + n0 + nn];
          mem[mm * D_MODEL + n0 + nn] = mo * (1.0f - g) + prop * g;
        }
        __syncthreads();
      }
    }
    grid_barrier(cnt, epoch, S2_NBLK);
  }
}

// ---------------------------------------------------------------------------
// Output projection: out = mem @ oW + ob  (8x768)
// ---------------------------------------------------------------------------
__global__ void out_proj(const float* __restrict__ mem,
                         const float* __restrict__ oW,
                         const float* __restrict__ ob,
                         float* __restrict__ out) {
  const int idx = blockIdx.x * 256 + threadIdx.x;
  if (idx >= BATCH * D_MODEL) return;
  const int bi = idx / D_MODEL, d = idx - bi * D_MODEL;
  float s = ob[d];
  const float* mr = mem + (size_t)bi * D_MODEL;
  for (int k = 0; k < D_MODEL; ++k) s += mr[k] * oW[(size_t)k * D_MODEL + d];
  out[idx] = s;
}

// ---------------------------------------------------------------------------
extern "C" void kernel_launch(void* const* d_in, const int* in_sizes, int n_in,
                              void* d_out, int out_size, void* d_ws,
                              size_t ws_size, hipStream_t stream) {
  const float* x   = (const float*)d_in[0];
  const float* qW  = (const float*)d_in[1];
  const float* qb  = (const float*)d_in[2];
  const float* kW  = (const float*)d_in[3];
  const float* kb  = (const float*)d_in[4];
  const float* f1W = (const float*)d_in[5];
  const float* f1b = (const float*)d_in[6];
  const float* f2W = (const float*)d_in[7];
  const float* f2b = (const float*)d_in[8];
  const float* lng = (const float*)d_in[9];
  const float* lnb = (const float*)d_in[10];
  const float* t1W = (const float*)d_in[11];
  const float* t1b = (const float*)d_in[12];
  const float* t2W = (const float*)d_in[13];
  const float* t2b = (const float*)d_in[14];
  const float* oW  = (const float*)d_in[15];
  const float* ob  = (const float*)d_in[16];
  float* out = (float*)d_out;

  float* ws     = (float*)d_ws;
  float* qbuf   = ws;                               // 16384*32
  float* kbuf   = qbuf + (size_t)ROWS * KQ_DIM;
  float* x1     = kbuf + (size_t)ROWS * KQ_DIM;
  float* y      = x1 + (size_t)ROWS * D_MODEL;
  float* a2     = y + (size_t)ROWS * D_MODEL;       // 8*1536
  float* mem    = a2 + (size_t)BATCH * D_FF;        // 8*768
  float* cntf   = mem + (size_t)BATCH * D_MODEL;    // 16 floats
  float* f1WT   = cntf + 16;                        // 1536 x 768
  float* f2WT   = f1WT + (size_t)D_FF * D_MODEL;    // 768  x 1536
  float* t1WT   = f2WT + (size_t)D_MODEL * D_FF;    // 1536 x 1536
  float* t2WT   = t1WT + (size_t)D_FF * D_FF;       // 768  x 1536
  unsigned* cnt = (unsigned*)cntf;

  // zero recurrent state + barrier counter (deterministic per call)
  hipMemsetAsync(mem, 0, ((size_t)BATCH * D_MODEL + 16) * sizeof(float), stream);

  // weight transposes (N x K copies for b64 B-fragment loads)
  transpose_kn<<<dim3(D_FF / 32, D_MODEL / 32), 256, 0, stream>>>(f1W, f1WT, D_MODEL, D_FF);
  transpose_kn<<<dim3(D_MODEL / 32, D_FF / 32), 256, 0, stream>>>(f2W, f2WT, D_FF, D_MODEL);
  transpose_kn<<<dim3(D_FF / 32, D_FF / 32), 256, 0, stream>>>(t1W, t1WT, D_FF, D_FF);
  transpose_kn<<<dim3(D_MODEL / 32, D_FF / 32), 256, 0, stream>>>(t2W, t2WT, D_FF, D_MODEL);

  router_proj<<<ROWS / 4, 256, 0, stream>>>(x, qW, qb, kW, kb, qbuf, kbuf);
  router_topk_gather<<<ROWS / 8, 256, 0, stream>>>(x, qbuf, kbuf, x1);

  const size_t s1_lds = (size_t)(D_FF * 16 + D_MODEL * 16) * sizeof(float);
  (void)hipFuncSetAttribute(reinterpret_cast<const void*>(system1_ffn),
                            hipFuncAttributeMaxDynamicSharedMemorySize,
                            (int)s1_lds);
  system1_ffn<<<ROWS / 16, 256, s1_lds, stream>>>(x1, f1WT, f1b, f2WT, f2b, y);
  layernorm_rows<<<ROWS / 8, 256, 0, stream>>>(y, lng, lnb);

  system2_scan<<<S2_NBLK, 256, 0, stream>>>(y, t1WT, t1b, t2WT, t2b, a2, mem, cnt);
  out_proj<<<(BATCH * D_MODEL + 255) / 256, 256, 0, stream>>>(mem, oW, ob, out);
}